// GCN_14972255993873
// MI455X (gfx1250) — compile-verified
//
#include <hip/hip_runtime.h>

typedef float v2f __attribute__((ext_vector_type(2)));
typedef float v8f __attribute__((ext_vector_type(8)));

#define N_NODES   100000
#define N_EDGES   800000
#define N_FEAT    128
#define HIDDEN    512
#define N_CLASSES 2
#define N_GRAPHS  512

// ---------------------------------------------------------------------------
// Degree / normalization
// ---------------------------------------------------------------------------
__global__ void deg_count_kernel(const long long* __restrict__ dst,
                                 unsigned* __restrict__ deg) {
    int e = blockIdx.x * blockDim.x + threadIdx.x;
    if (e < N_EDGES) atomicAdd(&deg[(int)dst[e]], 1u);
}

__global__ void dis_kernel(const unsigned* __restrict__ deg,
                           float* __restrict__ dis) {
    int i = blockIdx.x * blockDim.x + threadIdx.x;
    if (i < N_NODES) dis[i] = rsqrtf((float)deg[i] + 1.0f);  // +1 self loop
}

// ---------------------------------------------------------------------------
// WMMA fp32 GEMM: C[M x 512] = (relu?)(A[M x K]) @ W[K x 512]
// One wave computes a 32-row x 128-col slab: 2 M-tiles x 8 N-tiles of 16x16,
// so each B fragment is reused by two WMMAs (halves B-load issue pressure).
// A layout (16x4 f32): lane L (L&15 = M-row), lanes>=16 hold K+2/K+3.
// B layout (4x16 f32): lane L (L&15 = N-col), lanes>=16 hold rows K+2/K+3.
// ---------------------------------------------------------------------------
template <int K, bool RELU>
__global__ __launch_bounds__(256)
void gemm_wmma_f32(const float* __restrict__ A, const float* __restrict__ W,
                   float* __restrict__ C, int M) {
    const int wave = (blockIdx.x * blockDim.x + threadIdx.x) >> 5;
    const int lane = threadIdx.x & 31;
    const int colGroups = HIDDEN / 128;              // 4
    const int mt  = wave / colGroups;
    const int cg  = wave % colGroups;
    const int m0  = mt * 32;
    if (m0 >= M) return;
    const int n0   = cg * 128;
    const int half = lane >> 4;                      // 0 or 1
    const int l15  = lane & 15;

    v8f acc0[8] = {};                                // rows m0 .. m0+15
    v8f acc1[8] = {};                                // rows m0+16 .. m0+31

    const float* arow0 = A + (size_t)(m0 + l15) * K;
    const float* arow1 = A + (size_t)(m0 + 16 + l15) * K;
    for (int k0 = 0; k0 < K; k0 += 4) {
        const int kc = k0 + half * 2;
        v2f a0, a1;
        a0.x = arow0[kc];
        a0.y = arow0[kc + 1];
        a1.x = arow1[kc];
        a1.y = arow1[kc + 1];
        if (RELU) {
            a0.x = fmaxf(a0.x, 0.0f); a0.y = fmaxf(a0.y, 0.0f);
            a1.x = fmaxf(a1.x, 0.0f); a1.y = fmaxf(a1.y, 0.0f);
        }
        const float* w0 = W + (size_t)kc * HIDDEN + n0 + l15;
#pragma unroll
        for (int t = 0; t < 8; ++t) {
            v2f b;
            b.x = w0[t * 16];
            b.y = w0[t * 16 + HIDDEN];
            acc0[t] = __builtin_amdgcn_wmma_f32_16x16x4_f32(
                false, a0, false, b, (short)0, acc0[t], false, false);
            acc1[t] = __builtin_amdgcn_wmma_f32_16x16x4_f32(
                false, a1, false, b, (short)0, acc1[t], false, false);
        }
    }

    // C/D layout: VGPR r -> rows base+r (lanes 0-15) and base+r+8 (lanes 16-31)
#pragma unroll
    for (int t = 0; t < 8; ++t) {
        const int col = n0 + t * 16 + l15;
#pragma unroll
        for (int r = 0; r < 8; ++r) {
            const int row0 = m0 + r + half * 8;
            C[(size_t)row0 * HIDDEN + col] = acc0[t][r];
            C[(size_t)(row0 + 16) * HIDDEN + col] = acc1[t][r];
        }
    }
}

// ---------------------------------------------------------------------------
// out[i,h] = dis[i]^2 * xw[i,h] + bias[h]   (self-loop contribution + bias)
// ---------------------------------------------------------------------------
__global__ void selfloop_bias_kernel(const float* __restrict__ xw,
                                     const float* __restrict__ dis,
                                     const float* __restrict__ bias,
                                     float* __restrict__ out) {
    int idx = blockIdx.x * blockDim.x + threadIdx.x;
    if (idx < N_NODES * HIDDEN) {
        int node = idx >> 9;          // /512
        int h    = idx & (HIDDEN - 1);
        float s  = dis[node];
        out[idx] = s * s * xw[idx] + bias[h];
    }
}

// ---------------------------------------------------------------------------
// Edge scatter: out[d,:] += dis[s]*dis[d] * xw[s,:].  One wave per edge.
// ---------------------------------------------------------------------------
__global__ __launch_bounds__(256)
void edge_aggregate_kernel(const long long* __restrict__ src,
                           const long long* __restrict__ dst,
                           const float* __restrict__ xw,
                           const float* __restrict__ dis,
                           float* __restrict__ out) {
    const int e    = (blockIdx.x * blockDim.x + threadIdx.x) >> 5;
    const int lane = threadIdx.x & 31;
    if (e >= N_EDGES) return;
    const int s = (int)src[e];
    const int d = (int)dst[e];
    const float norm = dis[s] * dis[d];
    const float4* xr = (const float4*)(xw + (size_t)s * HIDDEN);
    float* orow = out + (size_t)d * HIDDEN;
#pragma unroll
    for (int i = 0; i < 4; ++i) {
        float4 v = xr[lane + i * 32];
        int base = (lane + i * 32) * 4;
        atomicAdd(&orow[base + 0], norm * v.x);
        atomicAdd(&orow[base + 1], norm * v.y);
        atomicAdd(&orow[base + 2], norm * v.z);
        atomicAdd(&orow[base + 3], norm * v.w);
    }
}

// ---------------------------------------------------------------------------
// Global mean pool
// ---------------------------------------------------------------------------
__global__ void pool_sum_kernel(const float* __restrict__ h,
                                const long long* __restrict__ batch,
                                float* __restrict__ gsum) {
    int idx = blockIdx.x * blockDim.x + threadIdx.x;
    if (idx < N_NODES * HIDDEN) {
        int node = idx >> 9;
        int f    = idx & (HIDDEN - 1);
        int g    = (int)batch[node];
        atomicAdd(&gsum[(size_t)g * HIDDEN + f], h[idx]);
    }
}

__global__ void pool_cnt_kernel(const long long* __restrict__ batch,
                                float* __restrict__ gcnt) {
    int n = blockIdx.x * blockDim.x + threadIdx.x;
    if (n < N_NODES) atomicAdd(&gcnt[(int)batch[n]], 1.0f);
}

// ---------------------------------------------------------------------------
// out[g,c] = (gsum[g,:]/cnt[g]) @ Wl[:,c] + bl[c]
// ---------------------------------------------------------------------------
__global__ void final_linear_kernel(const float* __restrict__ gsum,
                                    const float* __restrict__ gcnt,
                                    const float* __restrict__ Wl,
                                    const float* __restrict__ bl,
                                    float* __restrict__ out) {
    int idx = blockIdx.x * blockDim.x + threadIdx.x;
    if (idx < N_GRAPHS * N_CLASSES) {
        int g = idx / N_CLASSES;
        int c = idx % N_CLASSES;
        float inv = 1.0f / fmaxf(gcnt[g], 1.0f);
        float acc = 0.0f;
        for (int k = 0; k < HIDDEN; ++k)
            acc += gsum[(size_t)g * HIDDEN + k] * inv * Wl[k * N_CLASSES + c];
        out[idx] = acc + bl[c];
    }
}

// ---------------------------------------------------------------------------
extern "C" void kernel_launch(void* const* d_in, const int* in_sizes, int n_in,
                              void* d_out, int out_size, void* d_ws, size_t ws_size,
                              hipStream_t stream) {
    const float*     x     = (const float*)d_in[0];
    const long long* ei    = (const long long*)d_in[1];   // [2, N_EDGES] int64
    const long long* batch = (const long long*)d_in[2];   // [N_NODES] int64
    const float* W1 = (const float*)d_in[3];
    const float* b1 = (const float*)d_in[4];
    const float* W2 = (const float*)d_in[5];
    const float* b2 = (const float*)d_in[6];
    const float* W3 = (const float*)d_in[7];
    const float* b3 = (const float*)d_in[8];
    const float* Wl = (const float*)d_in[9];
    const float* bl = (const float*)d_in[10];
    float* out = (float*)d_out;

    const long long* srcp = ei;
    const long long* dstp = ei + N_EDGES;

    // Workspace layout
    const size_t NH = (size_t)N_NODES * HIDDEN;
    float*    buf1 = (float*)d_ws;            // h buffers (ping-pong)
    float*    buf2 = buf1 + NH;               // xw scratch
    float*    dis  = buf2 + NH;
    unsigned* deg  = (unsigned*)(dis + N_NODES);
    float*    gsum = (float*)(deg + N_NODES);
    float*    gcnt = gsum + (size_t)N_GRAPHS * HIDDEN;

    // Zero atomic accumulators every call (graph-capture-safe async memsets)
    hipMemsetAsync(deg,  0, N_NODES * sizeof(unsigned), stream);
    hipMemsetAsync(gsum, 0, ((size_t)N_GRAPHS * HIDDEN + N_GRAPHS) * sizeof(float), stream);

    // Normalization
    deg_count_kernel<<<(N_EDGES + 255) / 256, 256, 0, stream>>>(dstp, deg);
    dis_kernel<<<(N_NODES + 255) / 256, 256, 0, stream>>>(deg, dis);

    const int waves  = (N_NODES / 32) * (HIDDEN / 128);   // 12500 waves
    const int gblks  = (waves * 32 + 255) / 256;
    const int nhblks = (int)((NH + 255) / 256);
    const int eblks  = (int)(((size_t)N_EDGES * 32 + 255) / 256);

    // ----- Layer 1: xw = x @ W1 ; h1 = aggregate + b1  (ReLU fused into next GEMM load)
    gemm_wmma_f32<N_FEAT, false><<<gblks, 256, 0, stream>>>(x, W1, buf2, N_NODES);
    selfloop_bias_kernel<<<nhblks, 256, 0, stream>>>(buf2, dis, b1, buf1);
    edge_aggregate_kernel<<<eblks, 256, 0, stream>>>(srcp, dstp, buf2, dis, buf1);

    // ----- Layer 2: xw = relu(h1) @ W2 ; h2 = aggregate + b2
    gemm_wmma_f32<HIDDEN, true><<<gblks, 256, 0, stream>>>(buf1, W2, buf2, N_NODES);
    selfloop_bias_kernel<<<nhblks, 256, 0, stream>>>(buf2, dis, b2, buf1);
    edge_aggregate_kernel<<<eblks, 256, 0, stream>>>(srcp, dstp, buf2, dis, buf1);

    // ----- Layer 3: xw = relu(h2) @ W3 ; h3 = aggregate + b3 (no ReLU after)
    gemm_wmma_f32<HIDDEN, true><<<gblks, 256, 0, stream>>>(buf1, W3, buf2, N_NODES);
    selfloop_bias_kernel<<<nhblks, 256, 0, stream>>>(buf2, dis, b3, buf1);
    edge_aggregate_kernel<<<eblks, 256, 0, stream>>>(srcp, dstp, buf2, dis, buf1);

    // ----- Mean pool + final linear
    pool_sum_kernel<<<nhblks, 256, 0, stream>>>(buf1, batch, gsum);
    pool_cnt_kernel<<<(N_NODES + 255) / 256, 256, 0, stream>>>(batch, gcnt);
    final_linear_kernel<<<(N_GRAPHS * N_CLASSES + 255) / 256, 256, 0, stream>>>(
        gsum, gcnt, Wl, bl, out);
}